// Gaussians_60945585930483
// MI455X (gfx1250) — compile-verified
//
#include <hip/hip_runtime.h>
#include <hip/hip_bf16.h>
#include <stdint.h>

#define BLK 256

// Covariance from quaternion+scale for 3D Gaussians.
// Memory-bound streaming kernel using CDNA5 async global<->LDS DMA ops for the
// misaligned (12B / 36B per row) streams, and direct b128 loads for quaternions.
__global__ __launch_bounds__(BLK) void gauss_cov_kernel(
    const float* __restrict__ quat,
    const float* __restrict__ scale,
    float* __restrict__ out,
    int n)
{
    __shared__ __align__(16) float lds_s[3 * BLK];   // staged scales  (3 KB)
    __shared__ __align__(16) float lds_o[9 * BLK];   // staged output  (9 KB)

    const int tid = threadIdx.x;
    const long long base = (long long)blockIdx.x * BLK;
    const bool full = (base + BLK) <= (long long)n;

    float4 qv;
    float sx, sy, sz;

    if (full) {
        // --- async stage of scales: 768 floats = 192 x 16B, memory -> LDS direct
        const float* sp = scale + base * 3;               // uniform -> SGPR pair
        if (tid < (3 * BLK) / 4) {
            unsigned loff = (unsigned)(uintptr_t)(&lds_s[0]) + (unsigned)tid * 16u;
            unsigned goff = (unsigned)tid * 16u;
            asm volatile("global_load_async_to_lds_b128 %0, %1, %2"
                         :: "v"(loff), "v"(goff), "s"(sp) : "memory");
        }
        // quaternion rows are 16B and 16B-aligned: direct per-lane b128 load
        qv = reinterpret_cast<const float4*>(quat)[base + tid];
        asm volatile("s_wait_asynccnt 0" ::: "memory");
        __syncthreads();
        sx = lds_s[tid * 3 + 0];   // stride 3 mod 64 banks -> conflict-free
        sy = lds_s[tid * 3 + 1];
        sz = lds_s[tid * 3 + 2];
    } else {
        long long g = base + tid;
        if (g < n) {
            qv = reinterpret_cast<const float4*>(quat)[g];
            sx = scale[g * 3 + 0];
            sy = scale[g * 3 + 1];
            sz = scale[g * 3 + 2];
        } else {
            qv = make_float4(1.f, 0.f, 0.f, 0.f);
            sx = sy = sz = 0.f;
        }
    }

    // --- normalize quaternion: inv = 1 / max(||q||, 1e-12)
    float r = qv.x, x = qv.y, y = qv.z, z = qv.w;
    float n2  = fmaxf(r * r + x * x + y * y + z * z, 1e-24f);
    float inv = rsqrtf(n2);
    inv = inv * (1.5f - 0.5f * n2 * inv * inv);   // one Newton step: ~1 ulp
    r *= inv; x *= inv; y *= inv; z *= inv;

    // --- M = R * diag(s)
    float m00 = (1.f - 2.f * (y * y + z * z)) * sx;
    float m01 = (2.f * (x * y - r * z)) * sy;
    float m02 = (2.f * (x * z + r * y)) * sz;
    float m10 = (2.f * (x * y + r * z)) * sx;
    float m11 = (1.f - 2.f * (x * x + z * z)) * sy;
    float m12 = (2.f * (y * z - r * x)) * sz;
    float m20 = (2.f * (x * z - r * y)) * sx;
    float m21 = (2.f * (y * z + r * x)) * sy;
    float m22 = (1.f - 2.f * (x * x + y * y)) * sz;

    // --- C = M * M^T (symmetric)
    float c00 = m00 * m00 + m01 * m01 + m02 * m02;
    float c01 = m00 * m10 + m01 * m11 + m02 * m12;
    float c02 = m00 * m20 + m01 * m21 + m02 * m22;
    float c11 = m10 * m10 + m11 * m11 + m12 * m12;
    float c12 = m10 * m20 + m11 * m21 + m12 * m22;
    float c22 = m20 * m20 + m21 * m21 + m22 * m22;

    if (full) {
        // stride 9 mod 64 banks -> conflict-free LDS writes
        float* lo = &lds_o[tid * 9];
        lo[0] = c00; lo[1] = c01; lo[2] = c02;
        lo[3] = c01; lo[4] = c11; lo[5] = c12;
        lo[6] = c02; lo[7] = c12; lo[8] = c22;
        __syncthreads();

        // --- async drain of output: 2304 floats = 576 x 16B, LDS -> memory direct
        float* op = out + base * 9;                       // uniform -> SGPR pair
        unsigned lbase = (unsigned)(uintptr_t)(&lds_o[0]);
        {
            unsigned c0 = (unsigned)tid;
            unsigned g0 = c0 * 16u, l0 = lbase + c0 * 16u;
            asm volatile("global_store_async_from_lds_b128 %0, %1, %2"
                         :: "v"(g0), "v"(l0), "s"(op) : "memory");
            unsigned c1 = (unsigned)tid + BLK;
            unsigned g1 = c1 * 16u, l1 = lbase + c1 * 16u;
            asm volatile("global_store_async_from_lds_b128 %0, %1, %2"
                         :: "v"(g1), "v"(l1), "s"(op) : "memory");
            if (tid < (9 * BLK) / 4 - 2 * BLK) {          // 64 remaining chunks
                unsigned c2 = (unsigned)tid + 2 * BLK;
                unsigned g2 = c2 * 16u, l2 = lbase + c2 * 16u;
                asm volatile("global_store_async_from_lds_b128 %0, %1, %2"
                             :: "v"(g2), "v"(l2), "s"(op) : "memory");
            }
        }
        // ASYNCcnt drained by s_endpgm's implicit wait-idle.
    } else {
        long long g = base + tid;
        if (g < n) {
            float* op = out + g * 9;
            op[0] = c00; op[1] = c01; op[2] = c02;
            op[3] = c01; op[4] = c11; op[5] = c12;
            op[6] = c02; op[7] = c12; op[8] = c22;
        }
    }
}

extern "C" void kernel_launch(void* const* d_in, const int* in_sizes, int n_in,
                              void* d_out, int out_size, void* d_ws, size_t ws_size,
                              hipStream_t stream) {
    const float* quats  = (const float*)d_in[0];   // (N,4) float32
    const float* scales = (const float*)d_in[1];   // (N,3) float32
    float* out = (float*)d_out;                    // (N,3,3) float32
    int n = in_sizes[0] / 4;
    if (n <= 0) return;
    int grid = (n + BLK - 1) / BLK;
    gauss_cov_kernel<<<grid, BLK, 0, stream>>>(quats, scales, out, n);
}